// DOSLoss_30004641530405
// MI455X (gfx1250) — compile-verified
//
#include <hip/hip_runtime.h>
#include <math.h>

#define NEG_BIG (-1e30f)

typedef __attribute__((ext_vector_type(2))) float v2f;
typedef __attribute__((ext_vector_type(8))) float v8f;

// ---------------- init: zero the scalar accumulators ----------------
__global__ void init_kernel(float* acc) {
    acc[0] = 0.0f;   // f_loss
    acc[1] = 0.0f;   // g_loss
}

// ---------------- kernel 1: distances via WMMA reduction ----------------
// One wave (32 threads) per (batch b, 16-row k tile).
// Lane m   (m<16): squared-diff sums over d in [0,D/4) -> A[m,0], [D/4,D/2) -> A[m,1]
// Lane m+16     : d in [D/2,3D/4) -> A[m,2], [3D/4,D) -> A[m,3]
// D = A x ones(4x16): every column of D holds the rowsum -> 16 squared distances.
template <int DD>   // DD = compile-time D (0 = dynamic)
__global__ __launch_bounds__(32) void dist_kernel(
    const float* __restrict__ deep, const float* __restrict__ nbr,
    const float* __restrict__ w, const int* __restrict__ lengths,
    float* __restrict__ s_buf, float* __restrict__ acc,
    int B, int Kmax, int Drt) {
    const int D         = DD ? DD : Drt;
    const int lane      = threadIdx.x;          // 0..31, no divergence before WMMA
    const int tilesPerB = Kmax >> 4;
    const int b         = blockIdx.x / tilesPerB;
    const int k0        = (blockIdx.x % tilesPerB) << 4;
    const int m         = lane & 15;            // tile row
    const int row       = k0 + m;               // k index
    const int half      = lane >> 4;            // 0 or 1
    const int Dq        = D >> 2;               // chunk length
    const int dbase     = half * (D >> 1);

    const float* __restrict__ nrow = nbr + ((size_t)b * Kmax + row) * (size_t)D;
    const float* __restrict__ frow = deep + (size_t)b * D;

    float a0, a1;
    if ((D & 15) == 0) {                        // float4 path (uniform branch)
        const int Dq4 = Dq >> 2;                // float4s per chunk
        const float4* __restrict__ n4 = (const float4*)(nrow + dbase);
        const float4* __restrict__ f4 = (const float4*)(frow + dbase);
        float s0 = 0.f, s1 = 0.f, s2 = 0.f, s3 = 0.f;
        #pragma unroll 4
        for (int i = 0; i < Dq4; ++i) {
            float4 fv = f4[i], nv = n4[i];
            float dx = fv.x - nv.x, dy = fv.y - nv.y;
            float dz = fv.z - nv.z, dw = fv.w - nv.w;
            s0 = fmaf(dx, dx, s0); s1 = fmaf(dy, dy, s1);
            s2 = fmaf(dz, dz, s2); s3 = fmaf(dw, dw, s3);
        }
        a0 = (s0 + s1) + (s2 + s3);
        s0 = s1 = s2 = s3 = 0.f;
        #pragma unroll 4
        for (int i = 0; i < Dq4; ++i) {
            float4 fv = f4[Dq4 + i], nv = n4[Dq4 + i];
            float dx = fv.x - nv.x, dy = fv.y - nv.y;
            float dz = fv.z - nv.z, dw = fv.w - nv.w;
            s0 = fmaf(dx, dx, s0); s1 = fmaf(dy, dy, s1);
            s2 = fmaf(dz, dz, s2); s3 = fmaf(dw, dw, s3);
        }
        a1 = (s0 + s1) + (s2 + s3);
    } else {                                    // scalar fallback (uniform branch)
        a0 = 0.f; a1 = 0.f;
        for (int i = 0; i < Dq; ++i) {
            float d0 = frow[dbase + i] - nrow[dbase + i];
            a0 = fmaf(d0, d0, a0);
        }
        for (int i = 0; i < Dq; ++i) {
            float d1 = frow[dbase + Dq + i] - nrow[dbase + Dq + i];
            a1 = fmaf(d1, d1, a1);
        }
    }

    v2f a;    a[0] = a0;   a[1] = a1;
    v2f ones; ones[0] = 1.0f; ones[1] = 1.0f;
    v8f c = {};
    // EXEC is all-ones here (only uniform branches above).
    c = __builtin_amdgcn_wmma_f32_16x16x4_f32(
            /*neg_a=*/false, a, /*neg_b=*/false, ones,
            /*c_mod=*/(short)0, c, /*reuse_a=*/false, /*reuse_b=*/false);

    // Column N=0 of D lives in lane 0 (rows 0..7) and lane 16 (rows 8..15).
    __shared__ float red[16];
    if (lane == 0) {
        #pragma unroll
        for (int j = 0; j < 8; ++j) red[j] = c[j];
    } else if (lane == 16) {
        #pragma unroll
        for (int j = 0; j < 8; ++j) red[8 + j] = c[j];
    }
    __syncthreads();

    float contrib = 0.0f;
    if (lane < 16) {
        float dist  = sqrtf(red[lane]);
        int   len   = lengths[b];
        bool  valid = row < len;
        float sv    = -w[(size_t)b * Kmax + row] * dist;
        s_buf[(size_t)b * Kmax + row] = valid ? sv : NEG_BIG;
        contrib = valid ? sv : 0.0f;
    }
    #pragma unroll
    for (int off = 16; off > 0; off >>= 1)
        contrib += __shfl_down(contrib, off, 32);
    if (lane == 0) atomicAdd(&acc[0], contrib);
}

// ---------------- kernel 2: per-batch masked softmax over Kmax ----------------
__global__ __launch_bounds__(256) void softmax_kernel(
    const float* __restrict__ s_buf, float* __restrict__ rho, int Kmax) {
    const int b   = blockIdx.x;
    const int tid = threadIdx.x;
    __shared__ float sm[256];
    const float* __restrict__ srow = s_buf + (size_t)b * Kmax;
    float*       __restrict__ rrow = rho  + (size_t)b * Kmax;

    float mx = -INFINITY;
    for (int k = tid; k < Kmax; k += 256) mx = fmaxf(mx, srow[k]);
    sm[tid] = mx; __syncthreads();
    for (int s = 128; s > 0; s >>= 1) {
        if (tid < s) sm[tid] = fmaxf(sm[tid], sm[tid + s]);
        __syncthreads();
    }
    mx = sm[0]; __syncthreads();

    float sum = 0.0f;
    for (int k = tid; k < Kmax; k += 256) {
        float e = expf(srow[k] - mx);   // masked rows: exp(-1e30 - mx) == 0
        rrow[k] = e;
        sum += e;
    }
    sm[tid] = sum; __syncthreads();
    for (int s = 128; s > 0; s >>= 1) {
        if (tid < s) sm[tid] += sm[tid + s];
        __syncthreads();
    }
    float inv = 1.0f / sm[0];
    for (int k = tid; k < Kmax; k += 256) rrow[k] *= inv;
}

// ---------------- kernel 3: per-row two-pass logsumexp + g_loss ----------------
// One wave per (b,k) row of cls_score; skips rows with rho == 0 (exact: those
// contribute rho*ce == 0). Pass 1: row max (pure fmax). Pass 2: sum of exp.
// The 4KB row stays in L0/L2 between passes -> no extra HBM traffic.
template <int CC>   // CC = compile-time C (0 = dynamic)
__global__ __launch_bounds__(256) void ce_kernel(
    const float* __restrict__ cls, const float* __restrict__ rho,
    const int* __restrict__ target, float* __restrict__ acc,
    int Kmax, int Crt, int nrows) {
    const int C    = CC ? CC : Crt;
    const int tid  = threadIdx.x;
    const int lane = tid & 31;
    const int wid  = tid >> 5;                     // 0..7
    const int row  = blockIdx.x * 8 + wid;
    __shared__ float wsum[8];

    float contrib = 0.0f;
    if (row < nrows) {
        float r = rho[row];                        // wave-uniform
        if (r != 0.0f) {
            const int b = row / Kmax;
            const float* __restrict__ x = cls + (size_t)row * C;
            float m = -INFINITY;
            float s = 0.0f;
            if ((C & 3) == 0) {                    // float4 path, rows 16B-aligned
                const float4* __restrict__ x4 = (const float4*)x;
                const int C4 = C >> 2;
                // pass 1: max
                for (int i = lane; i < C4; i += 32) {
                    float4 v = x4[i];
                    m = fmaxf(m, fmaxf(fmaxf(v.x, v.y), fmaxf(v.z, v.w)));
                }
                #pragma unroll
                for (int off = 16; off > 0; off >>= 1)
                    m = fmaxf(m, __shfl_xor(m, off, 32));
                // pass 2: sum exp (row now cache-resident)
                float p0 = 0.f, p1 = 0.f, p2 = 0.f, p3 = 0.f;
                for (int i = lane; i < C4; i += 32) {
                    float4 v = x4[i];
                    p0 += expf(v.x - m); p1 += expf(v.y - m);
                    p2 += expf(v.z - m); p3 += expf(v.w - m);
                }
                s = (p0 + p1) + (p2 + p3);
            } else {                               // scalar fallback
                for (int i = lane; i < C; i += 32) m = fmaxf(m, x[i]);
                #pragma unroll
                for (int off = 16; off > 0; off >>= 1)
                    m = fmaxf(m, __shfl_xor(m, off, 32));
                for (int i = lane; i < C; i += 32) s += expf(x[i] - m);
            }
            #pragma unroll
            for (int off = 16; off > 0; off >>= 1)
                s += __shfl_xor(s, off, 32);
            if (lane == 0) {
                float lse = m + logf(s);
                float xt  = x[target[b]];
                contrib = r * (lse - xt);          // rho * cross-entropy
            }
        }
    }
    if (lane == 0) wsum[wid] = contrib;
    __syncthreads();
    if (tid == 0) {
        float t = 0.0f;
        #pragma unroll
        for (int j = 0; j < 8; ++j) t += wsum[j];
        if (t != 0.0f) atomicAdd(&acc[1], t);
    }
}

// ---------------- finalize ----------------
__global__ void fin_kernel(const float* __restrict__ acc, float* __restrict__ out) {
    out[0] = acc[0] + acc[1];
}

extern "C" void kernel_launch(void* const* d_in, const int* in_sizes, int n_in,
                              void* d_out, int out_size, void* d_ws, size_t ws_size,
                              hipStream_t stream) {
    const float* deep    = (const float*)d_in[0];   // [B,D]
    const float* nbr     = (const float*)d_in[1];   // [B,Kmax,D]
    const float* w       = (const float*)d_in[2];   // [B,Kmax]
    const float* cls     = (const float*)d_in[3];   // [B,Kmax,C]
    const int*   target  = (const int*)d_in[4];     // [B]
    const int*   lengths = (const int*)d_in[5];     // [B]
    float*       out     = (float*)d_out;

    const int B     = in_sizes[4];
    const int D     = in_sizes[0] / B;
    const int Kmax  = in_sizes[2] / B;
    const int C     = in_sizes[3] / (B * Kmax);
    const int nrows = B * Kmax;

    float* acc   = (float*)d_ws;                    // [0]=f_loss, [1]=g_loss
    float* s_buf = acc + 16;                        // [B*Kmax]
    float* rho   = s_buf + (size_t)nrows;           // [B*Kmax]

    init_kernel<<<1, 1, 0, stream>>>(acc);

    if (D == 256)
        dist_kernel<256><<<B * (Kmax / 16), 32, 0, stream>>>(
            deep, nbr, w, lengths, s_buf, acc, B, Kmax, D);
    else
        dist_kernel<0><<<B * (Kmax / 16), 32, 0, stream>>>(
            deep, nbr, w, lengths, s_buf, acc, B, Kmax, D);

    softmax_kernel<<<B, 256, 0, stream>>>(s_buf, rho, Kmax);

    if (C == 1000)
        ce_kernel<1000><<<(nrows + 7) / 8, 256, 0, stream>>>(
            cls, rho, target, acc, Kmax, C, nrows);
    else
        ce_kernel<0><<<(nrows + 7) / 8, 256, 0, stream>>>(
            cls, rho, target, acc, Kmax, C, nrows);

    fin_kernel<<<1, 1, 0, stream>>>(acc, out);
}